// TransportModule_45835890983689
// MI455X (gfx1250) — compile-verified
//
#include <hip/hip_runtime.h>

#define B_ 16
#define N_ 1024
#define D_ 64

typedef __attribute__((ext_vector_type(2))) float v2f;
typedef __attribute__((ext_vector_type(8))) float v8f;

__device__ __forceinline__ v8f wmma4(v2f a, v2f b, v8f c) {
    // V_WMMA_F32_16X16X4_F32 : D = A(16x4 f32) * B(4x16 f32) + C(16x16 f32)
    return __builtin_amdgcn_wmma_f32_16x16x4_f32(
        false, a, false, b, (short)0, c, false, false);
}

// ---------------------------------------------------------------- init
__global__ void ot_init(float* __restrict__ u, unsigned* __restrict__ minb) {
    int i = blockIdx.x * blockDim.x + threadIdx.x;
    if (i < B_ * N_) u[i] = 1.0f;            // exp(log_u = 0)
    if (i < B_)      minb[i] = 0x7F800000u;  // +inf bits
}

// ---------------------------------------------------------------- row norms
__global__ void ot_norms(const float* __restrict__ x, const float* __restrict__ y,
                         float* __restrict__ x2, float* __restrict__ y2) {
    int i = blockIdx.x * blockDim.x + threadIdx.x;  // 0 .. 2*B*N
    int total = B_ * N_;
    const float* src = (i < total) ? x : y;
    float* dst       = (i < total) ? x2 : y2;
    int r = (i < total) ? i : i - total;
    const float4* p = (const float4*)(src + (size_t)r * D_);
    float s = 0.f;
#pragma unroll
    for (int k = 0; k < D_ / 4; ++k) {
        float4 q = p[k];
        s += q.x * q.x + q.y * q.y + q.z * q.z + q.w * q.w;
    }
    dst[r] = s;
}

// ---------------------------------------------------------------- cost GEMM (WMMA f32)
// Each wave: 16 rows (n) x 64 cols (m). Block = 8 waves = 128 x 64 tile.
// Grid: b(16) * nBlk(8) * mStrip(16) = 2048 blocks.
__global__ void ot_cost(const float* __restrict__ X, const float* __restrict__ Y,
                        const float* __restrict__ x2, const float* __restrict__ y2,
                        float* __restrict__ K, unsigned* __restrict__ minb) {
    int bid    = blockIdx.x;
    int mStrip = bid & 15;
    int nBlk   = (bid >> 4) & 7;
    int b      = bid >> 7;
    int lane = threadIdx.x & 31;
    int wave = threadIdx.x >> 5;
    int lm   = lane & 15;
    int half = lane >> 4;
    int n0 = nBlk * 128 + wave * 16;
    int m0 = mStrip * 64;

    const float* Xrow  = X + ((size_t)(b * N_ + n0 + lm)) * D_ + half * 2;
    const float* Ybase = Y + ((size_t)(b * N_ + m0 + lm)) * D_ + half * 2;

    v8f acc0 = {}, acc1 = {}, acc2 = {}, acc3 = {};
#pragma unroll
    for (int kb = 0; kb < D_; kb += 4) {
        v2f a  = *(const v2f*)(Xrow + kb);
        v2f b0 = *(const v2f*)(Ybase + 0 * 16 * D_ + kb);
        v2f b1 = *(const v2f*)(Ybase + 1 * 16 * D_ + kb);
        v2f b2 = *(const v2f*)(Ybase + 2 * 16 * D_ + kb);
        v2f b3 = *(const v2f*)(Ybase + 3 * 16 * D_ + kb);
        acc0 = wmma4(a, b0, acc0);
        acc1 = wmma4(a, b1, acc1);
        acc2 = wmma4(a, b2, acc2);
        acc3 = wmma4(a, b3, acc3);
    }

    float mn = __uint_as_float(0x7F800000u);
#pragma unroll
    for (int r = 0; r < 8; ++r) {
        int row = n0 + half * 8 + r;          // C/D layout: lanes>=16 hold rows r+8
        float xn = x2[b * N_ + row];
        float av[4] = {acc0[r], acc1[r], acc2[r], acc3[r]};
#pragma unroll
        for (int t = 0; t < 4; ++t) {
            int col = m0 + t * 16 + lm;
            float c = xn + y2[b * N_ + col] - 2.0f * av[t];
            c = fmaxf(c, 0.0f);
            K[((size_t)(b * N_ + row)) * N_ + col] = c;
            mn = fminf(mn, c);
        }
    }
#pragma unroll
    for (int off = 16; off > 0; off >>= 1) mn = fminf(mn, __shfl_xor(mn, off, 32));
    if (lane == 0) atomicMin(&minb[b], __float_as_uint(mn));  // c>=0: bit order == value order
}

// ---------------------------------------------------------------- K = exp(-(C-Cmin)/eps)
__global__ void ot_expk(float* __restrict__ K, const unsigned* __restrict__ minb,
                        const int* __restrict__ epsp) {
    float inv_eps = 1.0f / (float)epsp[0];
    size_t idx    = (size_t)blockIdx.x * blockDim.x + threadIdx.x;
    size_t total  = (size_t)B_ * N_ * N_;
    size_t stride = (size_t)gridDim.x * blockDim.x;
    for (; idx < total; idx += stride) {
        int b = (int)(idx >> 20);  // N_*N_ = 1<<20
        float cmin = __uint_as_float(minb[b]);
        K[idx] = expf(-(K[idx] - cmin) * inv_eps);
    }
}

// ---------------------------------------------------------------- v = (1/N) / (K^T u)
// Block: 256 thr, batch b, 64 columns; 4 phase-waves split the n loop; LDS reduce.
__global__ void ot_colpass(const float* __restrict__ K, const float* __restrict__ u,
                           float* __restrict__ v) {
    __shared__ float ulds[N_];
    __shared__ float red[256];
    int b     = blockIdx.x >> 4;
    int col0  = (blockIdx.x & 15) * 64;
    for (int i = threadIdx.x; i < N_; i += 256) ulds[i] = u[b * N_ + i];
    __syncthreads();
    int phase = threadIdx.x >> 6;
    int c     = threadIdx.x & 63;
    const float* Kc = K + ((size_t)b * N_) * N_ + col0 + c;
    float t = 0.f;
    for (int i = 0; i < N_ / 4; ++i) {
        int n = 4 * i + phase;                 // each wave: fixed phase -> coalesced row read
        t += Kc[(size_t)n * N_] * ulds[n];
    }
    red[threadIdx.x] = t;
    __syncthreads();
    if (threadIdx.x < 64) {
        float tot = red[threadIdx.x] + red[threadIdx.x + 64] +
                    red[threadIdx.x + 128] + red[threadIdx.x + 192];
        v[b * N_ + col0 + threadIdx.x] = (1.0f / (float)N_) / tot;
    }
}

// ---------------------------------------------------------------- t = K v ; mode 0: u=(1/N)/t, mode 1: s=t
// Block: 256 thr = 8 waves, one row per wave. Grid: 16*128 = 2048.
__global__ void ot_rowpass(const float* __restrict__ K, const float* __restrict__ vv,
                           float* __restrict__ outv, int mode) {
    __shared__ float vlds[N_];
    int b      = blockIdx.x >> 7;
    int rowBlk = blockIdx.x & 127;
    for (int i = threadIdx.x; i < N_; i += 256) vlds[i] = vv[b * N_ + i];
    __syncthreads();
    int wave = threadIdx.x >> 5, lane = threadIdx.x & 31;
    int row = rowBlk * 8 + wave;
    const float* Kr = K + ((size_t)(b * N_ + row)) * N_;
    float t = 0.f;
#pragma unroll
    for (int i = 0; i < N_ / 128; ++i) {
        int m4 = lane + 32 * i;
        float4 q = ((const float4*)Kr)[m4];
        int m = m4 * 4;
        t += q.x * vlds[m] + q.y * vlds[m + 1] + q.z * vlds[m + 2] + q.w * vlds[m + 3];
    }
#pragma unroll
    for (int off = 16; off > 0; off >>= 1) t += __shfl_xor(t, off, 32);
    if (lane == 0) outv[b * N_ + row] = (mode == 0) ? ((1.0f / (float)N_) / t) : t;
}

// ---------------------------------------------------------------- out = K (v .* Y) / s   (WMMA f32)
// Wave: 16 rows x full d=64. Block = 8 waves. Grid: 16 b * 8 = 128.
__global__ void ot_outgemm(const float* __restrict__ K, const float* __restrict__ Y,
                           const float* __restrict__ v, const float* __restrict__ s,
                           float* __restrict__ out) {
    __shared__ float vlds[N_];
    int b    = blockIdx.x >> 3;
    int wave = threadIdx.x >> 5;
    int lane = threadIdx.x & 31;
    int lm = lane & 15, half = lane >> 4;
    int n0 = ((blockIdx.x & 7) * 8 + wave) * 16;
    for (int i = threadIdx.x; i < N_; i += 256) vlds[i] = v[b * N_ + i];
    __syncthreads();

    const float* Krow = K + ((size_t)(b * N_ + n0 + lm)) * N_ + half * 2;
    const float* Yb   = Y + (size_t)b * N_ * D_;
    v8f acc0 = {}, acc1 = {}, acc2 = {}, acc3 = {};
    for (int mb = 0; mb < N_; mb += 4) {
        v2f a = *(const v2f*)(Krow + mb);
        int mk = mb + half * 2;
        float v0 = vlds[mk], v1 = vlds[mk + 1];
        const float* yr0 = Yb + (size_t)mk * D_ + lm;
        const float* yr1 = yr0 + D_;
        v2f b0 = {v0 * yr0[0],  v1 * yr1[0]};
        v2f b1 = {v0 * yr0[16], v1 * yr1[16]};
        v2f b2 = {v0 * yr0[32], v1 * yr1[32]};
        v2f b3 = {v0 * yr0[48], v1 * yr1[48]};
        acc0 = wmma4(a, b0, acc0);
        acc1 = wmma4(a, b1, acc1);
        acc2 = wmma4(a, b2, acc2);
        acc3 = wmma4(a, b3, acc3);
    }
#pragma unroll
    for (int r = 0; r < 8; ++r) {
        int row = n0 + half * 8 + r;
        float rinv = 1.0f / fmaxf(s[b * N_ + row], 1e-12f);
        float* orow = out + ((size_t)(b * N_ + row)) * D_ + lm;
        orow[0]  = acc0[r] * rinv;
        orow[16] = acc1[r] * rinv;
        orow[32] = acc2[r] * rinv;
        orow[48] = acc3[r] * rinv;
    }
}

extern "C" void kernel_launch(void* const* d_in, const int* in_sizes, int n_in,
                              void* d_out, int out_size, void* d_ws, size_t ws_size,
                              hipStream_t stream) {
    const float* x  = (const float*)d_in[0];
    const float* y  = (const float*)d_in[1];
    const int* eps  = (const int*)d_in[2];
    float* out = (float*)d_out;
    float* ws  = (float*)d_ws;

    const size_t SZ_K = (size_t)B_ * N_ * N_;  // 16M floats = 64 MB (L2-resident)
    float* K  = ws;
    float* x2 = ws + SZ_K;
    float* y2 = x2 + B_ * N_;
    float* u  = y2 + B_ * N_;
    float* v  = u + B_ * N_;
    float* s  = v + B_ * N_;
    unsigned* minb = (unsigned*)(s + B_ * N_);

    ot_init<<<(B_ * N_ + 255) / 256, 256, 0, stream>>>(u, minb);
    ot_norms<<<(2 * B_ * N_) / 256, 256, 0, stream>>>(x, y, x2, y2);
    ot_cost<<<2048, 256, 0, stream>>>(x, y, x2, y2, K, minb);
    ot_expk<<<4096, 256, 0, stream>>>(K, minb, eps);

    for (int it = 0; it < 50; ++it) {
        ot_colpass<<<256, 256, 0, stream>>>(K, u, v);       // v = b / K^T u
        ot_rowpass<<<2048, 256, 0, stream>>>(K, v, u, 0);   // u = a / K v
    }
    ot_colpass<<<256, 256, 0, stream>>>(K, u, v);           // final v update
    ot_rowpass<<<2048, 256, 0, stream>>>(K, v, s, 1);       // rowsum s = K v
    ot_outgemm<<<128, 256, 0, stream>>>(K, y, v, s, out);   // out = K(v.*Y)/s
}